// TGCNCell_44951127720360
// MI455X (gfx1250) — compile-verified
//
#include <hip/hip_runtime.h>

#define NN 50000
#define BB 4
#define FF 16
#define HH 64
#define GI 80          // F + H
#define EE 800000
#define MTOT (NN * BB) // 200000 rows for the dense MLPs
#define ROWC (BB * GI) // 320 floats per x2/y2 row

typedef __attribute__((ext_vector_type(2))) float v2f;
typedef __attribute__((ext_vector_type(4))) float v4f;
typedef __attribute__((ext_vector_type(8))) float v8f;

// ---------------------------------------------------------------------------
// Kernel: pack xh2[R*80 + c]  (R = n*B + b), c<16 from x_t, else from h_prev
// ---------------------------------------------------------------------------
__global__ void pack_xh(const float* __restrict__ x_t,
                        const float* __restrict__ h_prev,
                        float* __restrict__ xh2) {
  int idx = blockIdx.x * blockDim.x + threadIdx.x; // over N*B*80 = 16M
  int c = idx % GI;
  int r = idx / GI;
  int b = r & (BB - 1);
  int n = r >> 2;
  float v = (c < FF) ? x_t[((long)b * NN + n) * FF + c]
                     : h_prev[((long)b * NN + n) * HH + (c - FF)];
  xh2[idx] = v;
}

// ---------------------------------------------------------------------------
// Kernel: pack weight matrix [K x 64] into paired layout:
//   Wp(v2f)[p*64 + n] = { W[2p][n], W[2p+1][n] }  -> one b64 per B fragment
// ---------------------------------------------------------------------------
__global__ void pack_w(const float* __restrict__ W, float* __restrict__ Wp,
                       int halfK) {
  int idx = blockIdx.x * blockDim.x + threadIdx.x; // halfK*64 threads
  if (idx >= halfK * HH) return;
  int p = idx / HH, n = idx % HH;
  v2f v;
  v.x = W[(2 * p) * HH + n];
  v.y = W[(2 * p + 1) * HH + n];
  *(v2f*)(Wp + 2 * idx) = v;
}

// ---------------------------------------------------------------------------
// CSR build: histogram -> single-block scan -> scatter
// ---------------------------------------------------------------------------
__global__ void hist_rows(const int* __restrict__ rows, int* __restrict__ cnt) {
  int e = blockIdx.x * blockDim.x + threadIdx.x;
  atomicAdd(cnt + rows[e], 1);
}

__global__ void __launch_bounds__(1024)
scan_rows(int* __restrict__ cnt /* in: counts, out: excl offsets */,
          int* __restrict__ offs /* [NN+1] */) {
  __shared__ int tmp[1024];
  __shared__ int carry_s;
  const int t = threadIdx.x;
  if (t == 0) carry_s = 0;
  __syncthreads();
  for (int base = 0; base < NN; base += 1024) {
    const int i = base + t;
    const int v = (i < NN) ? cnt[i] : 0;
    const int carry = carry_s;
    tmp[t] = v;
    __syncthreads();
    int run = v;
    for (int off = 1; off < 1024; off <<= 1) {
      int add = (t >= off) ? tmp[t - off] : 0;
      __syncthreads();
      run += add;
      tmp[t] = run;
      __syncthreads();
    }
    const int excl = carry + run - v;
    if (i < NN) {
      cnt[i] = excl;   // becomes scatter cursor
      offs[i] = excl;  // CSR row start
    }
    __syncthreads();
    if (t == 1023) carry_s = carry + tmp[1023];
    __syncthreads();
  }
  if (t == 0) offs[NN] = carry_s; // == EE
}

__global__ void scatter_edges(const int* __restrict__ rows,
                              const int* __restrict__ cols,
                              const float* __restrict__ vals,
                              int* __restrict__ cursor,
                              int* __restrict__ ecol,
                              float* __restrict__ eval_) {
  int e = blockIdx.x * blockDim.x + threadIdx.x;
  int pos = atomicAdd(cursor + rows[e], 1);
  ecol[pos] = cols[e];
  eval_[pos] = vals[e];
}

// ---------------------------------------------------------------------------
// Gather SpMM: one wave per output row, register accumulation, no atomics.
// Lane covers cols {lane*4..+3}, {128+lane*4..+3}, {256+lane*2..+1} of 320.
// x2 (64 MB) is L2-resident on the 192 MB L2; gathers are 128B-coalesced.
// ---------------------------------------------------------------------------
__global__ void __launch_bounds__(256)
spmm_csr(const int* __restrict__ offs, const int* __restrict__ ecol,
         const float* __restrict__ eval_, const float* __restrict__ x2,
         float* __restrict__ y2) {
  const int row = (blockIdx.x * blockDim.x + threadIdx.x) >> 5;
  const int lane = threadIdx.x & 31;
  const int s = offs[row];
  const int e = offs[row + 1];
  const int c0 = lane * 4;
  const int c1 = 128 + lane * 4;
  const int c2 = 256 + lane * 2;
  v4f a0 = {}, a1 = {};
  v2f a2 = {};
  for (int p = s; p < e; ++p) {
    const int col = ecol[p];
    const float v = eval_[p];
    const float* src = x2 + (long)col * ROWC;
    v4f d0 = *(const v4f*)(src + c0);
    v4f d1 = *(const v4f*)(src + c1);
    v2f d2 = *(const v2f*)(src + c2);
    a0 += d0 * v;
    a1 += d1 * v;
    a2 += d2 * v;
  }
  float* dst = y2 + (long)row * ROWC;
  *(v4f*)(dst + c0) = a0;
  *(v4f*)(dst + c1) = a1;
  *(v2f*)(dst + c2) = a2;
}

// ---------------------------------------------------------------------------
// Wave-level MLP on a 16-row block:  pre = relu(A@W1 + b1) @ W2 + b2
// A (16x80 f32) preloaded in 20 v2f regs (16x16x4 A layout).
// B fragments: single b64 loads from pre-paired weights, 20 (16) fragments
// preloaded per column tile -> wmma chain issues back-to-back.
// h1 bounced through per-wave LDS to convert C-layout -> A-layout.
// ---------------------------------------------------------------------------
__device__ __forceinline__ void mlp16(const v2f* __restrict__ a, // [20]
                                      const v2f* __restrict__ W1p,
                                      const float* __restrict__ b1,
                                      const v2f* __restrict__ W2p,
                                      const float* __restrict__ b2,
                                      float* __restrict__ h1s, // per-wave 16*64
                                      int lane, v8f out[4]) {
  const int nc = lane & 15;
  const int kh = lane >> 4; // 0 or 1
  const int mrow = lane & 15;

  // ---- GEMM1: 16x80 @ 80x64 ----
#pragma unroll
  for (int t = 0; t < 4; ++t) {
    const int n = t * 16 + nc;
    v2f bv[20];
#pragma unroll
    for (int k = 0; k < 20; ++k) bv[k] = W1p[(2 * k + kh) * HH + n];
    v8f acc = {};
#pragma unroll
    for (int k = 0; k < 20; ++k)
      acc = __builtin_amdgcn_wmma_f32_16x16x4_f32(false, a[k], false, bv[k],
                                                  (short)0, acc, false, false);
    const float bias = b1[n];
#pragma unroll
    for (int i = 0; i < 8; ++i) {
      float x = acc[i] + bias;
      x = x > 0.f ? x : 0.f; // ReLU
      h1s[(i + 8 * kh) * HH + n] = x;
    }
  }

  // ---- reload h1 in A layout ----
  v2f a2[16];
#pragma unroll
  for (int k = 0; k < 16; ++k)
    a2[k] = *(const v2f*)(h1s + mrow * HH + 4 * k + 2 * kh);

  // ---- GEMM2: 16x64 @ 64x64 ----
#pragma unroll
  for (int t = 0; t < 4; ++t) {
    const int n = t * 16 + nc;
    v2f bv[16];
#pragma unroll
    for (int k = 0; k < 16; ++k) bv[k] = W2p[(2 * k + kh) * HH + n];
    v8f acc = {};
#pragma unroll
    for (int k = 0; k < 16; ++k)
      acc = __builtin_amdgcn_wmma_f32_16x16x4_f32(false, a2[k], false, bv[k],
                                                  (short)0, acc, false, false);
    const float bias = b2[n];
#pragma unroll
    for (int i = 0; i < 8; ++i) acc[i] += bias;
    out[t] = acc;
  }
}

__device__ __forceinline__ float sigmoidf_(float x) {
  return 1.f / (1.f + __expf(-x));
}

// ---------------------------------------------------------------------------
// z and r gates from shared y2; writes sigmoid(z) to z2 and r*h_prev in place
// into xh2's h-columns (x-columns already correct).
// ---------------------------------------------------------------------------
__global__ void __launch_bounds__(128)
gate_zr(const float* __restrict__ y2,
        const float* __restrict__ Wz1p, const float* __restrict__ bz1,
        const float* __restrict__ Wz2p, const float* __restrict__ bz2,
        const float* __restrict__ Wr1p, const float* __restrict__ br1,
        const float* __restrict__ Wr2p, const float* __restrict__ br2,
        const float* __restrict__ h_prev,
        float* __restrict__ z2, float* __restrict__ xh2) {
  __shared__ float h1s_all[4 * 16 * HH];
  const int wv = threadIdx.x >> 5;
  const int lane = threadIdx.x & 31;
  float* h1s = h1s_all + wv * 16 * HH;
  const int row0 = (blockIdx.x * 4 + wv) * 16;
  const int mrow = lane & 15;
  const int kh = lane >> 4;

  // preload A tile (16 rows x 80 K) — reused for both gates, all col tiles
  const float* arow = y2 + (long)(row0 + mrow) * GI;
  v2f a[20];
#pragma unroll
  for (int k = 0; k < 20; ++k) a[k] = *(const v2f*)(arow + 4 * k + 2 * kh);

  v8f pre[4];

  // ---- z gate ----
  mlp16(a, (const v2f*)Wz1p, bz1, (const v2f*)Wz2p, bz2, h1s, lane, pre);
#pragma unroll
  for (int t = 0; t < 4; ++t) {
    const int n = t * 16 + (lane & 15);
#pragma unroll
    for (int i = 0; i < 8; ++i) {
      const long R = row0 + i + 8 * kh;
      z2[R * HH + n] = sigmoidf_(pre[t][i]);
    }
  }

  // ---- r gate, fused with r*h_prev repack ----
  mlp16(a, (const v2f*)Wr1p, br1, (const v2f*)Wr2p, br2, h1s, lane, pre);
#pragma unroll
  for (int t = 0; t < 4; ++t) {
    const int n = t * 16 + (lane & 15);
#pragma unroll
    for (int i = 0; i < 8; ++i) {
      const long R = row0 + i + 8 * kh;
      const int b = (int)(R & 3);
      const int nn = (int)(R >> 2);
      const float rv = sigmoidf_(pre[t][i]);
      const float hp = h_prev[((long)b * NN + nn) * HH + n];
      xh2[R * GI + FF + n] = rv * hp;
    }
  }
}

// ---------------------------------------------------------------------------
// h-gate + final blend:  out = (1-z)*h_prev + z*tanh(mlp(y2))
// ---------------------------------------------------------------------------
__global__ void __launch_bounds__(128)
gate_h(const float* __restrict__ y2,
       const float* __restrict__ Wh1p, const float* __restrict__ bh1,
       const float* __restrict__ Wh2p, const float* __restrict__ bh2,
       const float* __restrict__ h_prev, const float* __restrict__ z2,
       float* __restrict__ out) {
  __shared__ float h1s_all[4 * 16 * HH];
  const int wv = threadIdx.x >> 5;
  const int lane = threadIdx.x & 31;
  float* h1s = h1s_all + wv * 16 * HH;
  const int row0 = (blockIdx.x * 4 + wv) * 16;
  const int mrow = lane & 15;
  const int kh = lane >> 4;

  const float* arow = y2 + (long)(row0 + mrow) * GI;
  v2f a[20];
#pragma unroll
  for (int k = 0; k < 20; ++k) a[k] = *(const v2f*)(arow + 4 * k + 2 * kh);

  v8f pre[4];
  mlp16(a, (const v2f*)Wh1p, bh1, (const v2f*)Wh2p, bh2, h1s, lane, pre);

#pragma unroll
  for (int t = 0; t < 4; ++t) {
    const int n = t * 16 + (lane & 15);
#pragma unroll
    for (int i = 0; i < 8; ++i) {
      const long R = row0 + i + 8 * kh;
      const int b = (int)(R & 3);
      const int nn = (int)(R >> 2);
      const float ht = tanhf(pre[t][i]);
      const float hp = h_prev[((long)b * NN + nn) * HH + n];
      const float zv = z2[R * HH + n];
      out[((long)b * NN + nn) * HH + n] = (1.f - zv) * hp + zv * ht;
    }
  }
}

// ---------------------------------------------------------------------------
// Launch
// ---------------------------------------------------------------------------
extern "C" void kernel_launch(void* const* d_in, const int* in_sizes, int n_in,
                              void* d_out, int out_size, void* d_ws,
                              size_t ws_size, hipStream_t stream) {
  const float* x_t    = (const float*)d_in[0];
  const float* h_prev = (const float*)d_in[1];
  const float* A_val  = (const float*)d_in[2];
  const int*   A_idx  = (const int*)d_in[3];
  const float* Wz1 = (const float*)d_in[4];  const float* bz1 = (const float*)d_in[5];
  const float* Wz2 = (const float*)d_in[6];  const float* bz2 = (const float*)d_in[7];
  const float* Wr1 = (const float*)d_in[8];  const float* br1 = (const float*)d_in[9];
  const float* Wr2 = (const float*)d_in[10]; const float* br2 = (const float*)d_in[11];
  const float* Wh1 = (const float*)d_in[12]; const float* bh1 = (const float*)d_in[13];
  const float* Wh2 = (const float*)d_in[14]; const float* bh2 = (const float*)d_in[15];
  float* out = (float*)d_out;

  const int* rows = A_idx;      // A_idx[0]
  const int* cols = A_idx + EE; // A_idx[1]

  // -------- workspace layout (256B aligned chunks) --------
  char* ws = (char*)d_ws;
  size_t off = 0;
  auto alloc = [&](size_t bytes) {
    char* p = ws + off;
    off += (bytes + 255) & ~(size_t)255;
    return p;
  };
  const size_t xh_bytes = (size_t)MTOT * GI * sizeof(float); // 64 MB
  float* xh2   = (float*)alloc(xh_bytes);
  float* y2    = (float*)alloc(xh_bytes);
  float* z2    = (float*)alloc((size_t)MTOT * HH * sizeof(float)); // 51.2 MB
  int*   offs  = (int*)alloc((NN + 1) * sizeof(int));
  int*   curs  = (int*)alloc(NN * sizeof(int));
  int*   ecol  = (int*)alloc(EE * sizeof(int));
  float* eval_ = (float*)alloc(EE * sizeof(float));
  float* Wz1p  = (float*)alloc(GI * HH * sizeof(float));
  float* Wz2p  = (float*)alloc(HH * HH * sizeof(float));
  float* Wr1p  = (float*)alloc(GI * HH * sizeof(float));
  float* Wr2p  = (float*)alloc(HH * HH * sizeof(float));
  float* Wh1p  = (float*)alloc(GI * HH * sizeof(float));
  float* Wh2p  = (float*)alloc(HH * HH * sizeof(float));

  // 1) pack concat(x_t, h_prev) into [N*B, 80]
  pack_xh<<<(MTOT * GI) / 256, 256, 0, stream>>>(x_t, h_prev, xh2);

  // 2) pack all 6 weight matrices into paired-B layout
  pack_w<<<(40 * HH) / 256, 256, 0, stream>>>(Wz1, Wz1p, 40);
  pack_w<<<(32 * HH) / 256, 256, 0, stream>>>(Wz2, Wz2p, 32);
  pack_w<<<(40 * HH) / 256, 256, 0, stream>>>(Wr1, Wr1p, 40);
  pack_w<<<(32 * HH) / 256, 256, 0, stream>>>(Wr2, Wr2p, 32);
  pack_w<<<(40 * HH) / 256, 256, 0, stream>>>(Wh1, Wh1p, 40);
  pack_w<<<(32 * HH) / 256, 256, 0, stream>>>(Wh2, Wh2p, 32);

  // 3) CSR build (per call; tiny vs. the 512M atomics it eliminates)
  hipMemsetAsync(curs, 0, NN * sizeof(int), stream);
  hist_rows<<<EE / 256, 256, 0, stream>>>(rows, curs);
  scan_rows<<<1, 1024, 0, stream>>>(curs, offs);
  scatter_edges<<<EE / 256, 256, 0, stream>>>(rows, cols, A_val, curs, ecol,
                                              eval_);

  // 4) y2 = A @ xh2   (gather SpMM, full overwrite, no atomics/memset)
  spmm_csr<<<(NN * 32) / 256, 256, 0, stream>>>(offs, ecol, eval_, xh2, y2);

  // 5) z, r gates; writes r*h_prev back into xh2's h-columns
  gate_zr<<<MTOT / 64, 128, 0, stream>>>(y2, Wz1p, bz1, Wz2p, bz2, Wr1p, br1,
                                         Wr2p, br2, h_prev, z2, xh2);

  // 6) y2 = A @ concat(x_t, r*h_prev)
  spmm_csr<<<(NN * 32) / 256, 256, 0, stream>>>(offs, ecol, eval_, xh2, y2);

  // 7) h-gate + blend
  gate_h<<<MTOT / 64, 128, 0, stream>>>(y2, Wh1p, bh1, Wh2p, bh2, h_prev, z2,
                                        out);
}